// DecoderRNN_25237227832024
// MI455X (gfx1250) — compile-verified
//
#include <hip/hip_runtime.h>

#define B 64
#define T 20
#define P 196
#define ENC 2048
#define E 512
#define H 512
#define A_DIM 512
#define V 10000
#define KCAT 3072   // ENC + E + H
#define G4H 2048    // 4*H

#define KC 64          // K-chunk staged in LDS per iteration (all K's are multiples of 64)
#define LDS_STRIDE 68  // floats: rows 272 B apart -> 16B-aligned b128 stores, bank-spread ds reads

typedef float v2f __attribute__((ext_vector_type(2)));
typedef float v8f __attribute__((ext_vector_type(8)));

// ---------------------------------------------------------------- utilities
__global__ void zero_kernel(float* __restrict__ p, int n) {
    int i = blockIdx.x * blockDim.x + threadIdx.x;
    if (i < n) p[i] = 0.f;
}

__global__ void emb_gather(const float* __restrict__ embW, const int* __restrict__ caps,
                           float* __restrict__ emb) {
    int i = blockIdx.x * blockDim.x + threadIdx.x;          // over B*T*E
    if (i >= B * T * E) return;
    int e  = i % E;
    int bt = i / E;
    emb[i] = embW[(size_t)caps[bt] * E + e];
}

// Wcat[n, 0:2048]=W_ih[:, :ENC] (ctx), [2048:2560]=W_ih[:, ENC:] (emb), [2560:3072]=W_hh
__global__ void build_wcat(const float* __restrict__ Wih, const float* __restrict__ Whh,
                           const float* __restrict__ bih, const float* __restrict__ bhh,
                           float* __restrict__ Wcat, float* __restrict__ bcat) {
    int i = blockIdx.x * blockDim.x + threadIdx.x;          // over G4H*KCAT
    if (i >= G4H * KCAT) return;
    int k = i % KCAT, n = i / KCAT;
    float v = (k < ENC + E) ? Wih[(size_t)n * (ENC + E) + k]
                            : Whh[(size_t)n * H + (k - (ENC + E))];
    Wcat[i] = v;
    if (i < G4H) bcat[i] = bih[i] + bhh[i];
}

// ------------------------------------------------ FP32 WMMA GEMM: C = A @ W^T + bias
// A:[M,K] row-major, W:[N,K] row-major, C row stride ldc. M%16==0, N%16==0, K%64==0.
// Grid: (ceil(nGroups/8), mTiles), 256 threads (8 waves). All waves of a block share one
// 16-row M-tile of A, which is staged global->LDS with async-to-LDS copies, double-buffered.
// Each wave computes a 16x64 tile (4 n-subtiles); ragged N is clamped (no early returns,
// EXEC all-ones at every WMMA, every wave reaches every barrier/async op).
__global__ void __launch_bounds__(256)
gemm_wmma(const float* __restrict__ Amat, const float* __restrict__ Wmat,
          const float* __restrict__ bias, float* __restrict__ C,
          int M, int N, int K, int ldc) {
    __shared__ float sA[2][16 * LDS_STRIDE];

    const int tid  = threadIdx.x;
    const int lane = tid & 31;
    const int nTiles  = N >> 4;
    const int nGroups = (nTiles + 3) >> 2;
    const int mt = blockIdx.y;

    int ng = blockIdx.x * 8 + (tid >> 5);
    const bool active = ng < nGroups;
    if (!active) ng = nGroups - 1;            // clamp: redundant compute, stores suppressed

    const int ln   = lane & 15;
    const int koff = (lane >> 4) << 1;        // lanes 0-15 -> K=k,k+1 ; 16-31 -> K=k+2,k+3

    const int t0 = ng * 4;
    const bool vt1 = (t0 + 1) < nTiles, vt2 = (t0 + 2) < nTiles, vt3 = (t0 + 3) < nTiles;
    const float* W0 = Wmat + (size_t)((t0 + 0) * 16 + ln) * K;
    const float* W1 = vt1 ? Wmat + (size_t)((t0 + 1) * 16 + ln) * K : W0;
    const float* W2 = vt2 ? Wmat + (size_t)((t0 + 2) * 16 + ln) * K : W0;
    const float* W3 = vt3 ? Wmat + (size_t)((t0 + 3) * 16 + ln) * K : W0;

    // --- async A staging: thread (row = tid>>4, seg = tid&15) copies one 16B segment/chunk
    const int srow = tid >> 4;   // 0..15
    const int sseg = tid & 15;   // 0..15 (16 segments x 4 floats = 64 floats per row)
    const unsigned long long gsrc =
        (unsigned long long)(uintptr_t)(Amat + (size_t)(mt * 16 + srow) * K + sseg * 4);
    unsigned int ldsDst[2];
    ldsDst[0] = (unsigned int)(uintptr_t)&sA[0][srow * LDS_STRIDE + sseg * 4];
    ldsDst[1] = (unsigned int)(uintptr_t)&sA[1][srow * LDS_STRIDE + sseg * 4];

    v8f acc0 = {}, acc1 = {}, acc2 = {}, acc3 = {};

    // prologue: stage chunk 0 into buffer 0
    {
        unsigned long long ga = gsrc;
        asm volatile("global_load_async_to_lds_b128 %0, %1, off"
                     :: "v"(ldsDst[0]), "v"(ga) : "memory");
    }

    int cur = 0;
    for (int kc = 0; kc < K; kc += KC) {
        asm volatile("s_wait_asynccnt 0" ::: "memory");   // chunk kc landed (this wave's copies)
        __syncthreads();                                   // ...and every wave's copies

        if (kc + KC < K) {                                 // stage next chunk into other buffer
            unsigned long long ga = gsrc + (unsigned long long)(kc + KC) * 4ull;
            asm volatile("global_load_async_to_lds_b128 %0, %1, off"
                         :: "v"(ldsDst[cur ^ 1]), "v"(ga) : "memory");
        }

        const float* __restrict__ bufp = sA[cur];
        #pragma unroll 4
        for (int k = 0; k < KC; k += 4) {
            v2f a = *(const v2f*)&bufp[ln * LDS_STRIDE + k + koff];   // ds_load_b64
            const int kk = kc + k + koff;
            v2f b0; b0.x = W0[kk]; b0.y = W0[kk + 1];
            v2f b1; b1.x = W1[kk]; b1.y = W1[kk + 1];
            v2f b2; b2.x = W2[kk]; b2.y = W2[kk + 1];
            v2f b3; b3.x = W3[kk]; b3.y = W3[kk + 1];
            acc0 = __builtin_amdgcn_wmma_f32_16x16x4_f32(false, a, false, b0, (short)0, acc0, false, false);
            acc1 = __builtin_amdgcn_wmma_f32_16x16x4_f32(false, a, false, b1, (short)0, acc1, false, false);
            acc2 = __builtin_amdgcn_wmma_f32_16x16x4_f32(false, a, false, b2, (short)0, acc2, false, false);
            acc3 = __builtin_amdgcn_wmma_f32_16x16x4_f32(false, a, false, b3, (short)0, acc3, false, false);
        }
        cur ^= 1;
    }

    // C/D layout: VGPR r -> lanes0-15: M=mt*16+r, lanes16-31: M=mt*16+8+r; N = tile*16 + ln
    if (active) {
        const int mbase = mt * 16 + ((lane >> 4) << 3);
        {
            int n0 = (t0 + 0) * 16 + ln;
            float bv = bias[n0];
            #pragma unroll
            for (int r = 0; r < 8; ++r) C[(size_t)(mbase + r) * ldc + n0] = acc0[r] + bv;
        }
        if (vt1) {
            int n0 = (t0 + 1) * 16 + ln;
            float bv = bias[n0];
            #pragma unroll
            for (int r = 0; r < 8; ++r) C[(size_t)(mbase + r) * ldc + n0] = acc1[r] + bv;
        }
        if (vt2) {
            int n0 = (t0 + 2) * 16 + ln;
            float bv = bias[n0];
            #pragma unroll
            for (int r = 0; r < 8; ++r) C[(size_t)(mbase + r) * ldc + n0] = acc2[r] + bv;
        }
        if (vt3) {
            int n0 = (t0 + 3) * 16 + ln;
            float bv = bias[n0];
            #pragma unroll
            for (int r = 0; r < 8; ++r) C[(size_t)(mbase + r) * ldc + n0] = acc3[r] + bv;
        }
    }
}

// ------------------------------------------- e[b,p] = wf . relu(att1[b,p,:]+att2[b,:]) + bf
__global__ void attn_score(const float* __restrict__ att1, const float* __restrict__ att2,
                           const float* __restrict__ wf, const float* __restrict__ bf,
                           float* __restrict__ ebuf) {
    int wave = blockIdx.x * (blockDim.x >> 5) + (threadIdx.x >> 5);   // one wave per (b,p)
    int lane = threadIdx.x & 31;
    if (wave >= B * P) return;
    int b = wave / P;
    const float* row = att1 + (size_t)wave * A_DIM;
    const float* a2  = att2 + (size_t)b * A_DIM;
    float s = 0.f;
    for (int a = lane; a < A_DIM; a += 32) {
        float v = row[a] + a2[a];
        s += (v > 0.f ? v : 0.f) * wf[a];
    }
    #pragma unroll
    for (int off = 16; off > 0; off >>= 1) s += __shfl_xor(s, off, 32);
    if (lane == 0) ebuf[wave] = s + bf[0];
}

// ---------------- softmax over P, write alphas[b,t,:], ctx = alpha.feats -> x[b, 0:ENC]
__global__ void softmax_ctx(const float* __restrict__ ebuf, const float* __restrict__ feats,
                            float* __restrict__ xbuf, float* __restrict__ outAlpha, int t) {
    __shared__ float sAl[P];
    __shared__ float sRed[256];
    int b = blockIdx.x, tid = threadIdx.x;
    const float* e = ebuf + b * P;

    float m = -3.402823e38f;
    for (int p = tid; p < P; p += 256) m = fmaxf(m, e[p]);
    sRed[tid] = m; __syncthreads();
    for (int s = 128; s > 0; s >>= 1) {
        if (tid < s) sRed[tid] = fmaxf(sRed[tid], sRed[tid + s]);
        __syncthreads();
    }
    m = sRed[0]; __syncthreads();

    float sum = 0.f;
    for (int p = tid; p < P; p += 256) { float ev = __expf(e[p] - m); sAl[p] = ev; sum += ev; }
    sRed[tid] = sum; __syncthreads();
    for (int s = 128; s > 0; s >>= 1) {
        if (tid < s) sRed[tid] += sRed[tid + s];
        __syncthreads();
    }
    float inv = 1.f / sRed[0]; __syncthreads();

    for (int p = tid; p < P; p += 256) {
        float a = sAl[p] * inv;
        sAl[p] = a;
        outAlpha[(size_t)b * T * P + (size_t)t * P + p] = a;
    }
    __syncthreads();

    const float* fb = feats + (size_t)b * P * ENC;
    for (int d = tid; d < ENC; d += 256) {
        float acc = 0.f;
        for (int p = 0; p < P; ++p) acc += sAl[p] * fb[(size_t)p * ENC + d];
        xbuf[(size_t)b * KCAT + d] = acc;
    }
}

// ----------------------------- x[b, ENC:ENC+E] = emb[b,t,:], x[b, ENC+E:] = h_prev[b,:]
__global__ void xfill(const float* __restrict__ emb, const float* __restrict__ h,
                      float* __restrict__ xbuf, int t) {
    int i = blockIdx.x * blockDim.x + threadIdx.x;   // over B*(E+H)
    if (i >= B * (E + H)) return;
    int b = i / (E + H), j = i % (E + H);
    float v = (j < E) ? emb[(size_t)b * T * E + (size_t)t * E + j]
                      : h[(size_t)b * H + (j - E)];
    xbuf[(size_t)b * KCAT + ENC + j] = v;
}

// -------------------------------------------------------------- LSTM pointwise update
__global__ void lstm_cell(const float* __restrict__ gates, float* __restrict__ h,
                          float* __restrict__ c) {
    int i = blockIdx.x * blockDim.x + threadIdx.x;   // over B*H
    if (i >= B * H) return;
    int b = i / H, j = i % H;
    const float* g = gates + (size_t)b * G4H;
    float ig = g[j], fg = g[H + j], gg = g[2 * H + j], og = g[3 * H + j];
    float si = 1.f / (1.f + __expf(-ig));
    float sf = 1.f / (1.f + __expf(-fg));
    float so = 1.f / (1.f + __expf(-og));
    float cn = sf * c[i] + si * tanhf(gg);
    c[i] = cn;
    h[i] = so * tanhf(cn);
}

// =============================================================== host-side launcher
extern "C" void kernel_launch(void* const* d_in, const int* in_sizes, int n_in,
                              void* d_out, int out_size, void* d_ws, size_t ws_size,
                              hipStream_t stream) {
    const float* feats = (const float*)d_in[0];     // [B,P,ENC]
    const int*   caps  = (const int*)d_in[1];       // [B,T]
    // d_in[2] caption_lengths: sort+unsort cancels, lengths never mask -> unused
    const float* embW  = (const float*)d_in[3];     // [V,E]
    const float* We    = (const float*)d_in[4];     // [A,ENC]
    const float* be    = (const float*)d_in[5];
    const float* Wd    = (const float*)d_in[6];     // [A,H]
    const float* bd    = (const float*)d_in[7];
    const float* wf    = (const float*)d_in[8];     // [1,A]
    const float* bf    = (const float*)d_in[9];
    const float* Wih   = (const float*)d_in[10];    // [4H, E+ENC]
    const float* Whh   = (const float*)d_in[11];    // [4H, H]
    const float* bih   = (const float*)d_in[12];
    const float* bhh   = (const float*)d_in[13];
    const float* fcW   = (const float*)d_in[14];    // [V,H]
    const float* fcb   = (const float*)d_in[15];
    (void)in_sizes; (void)n_in; (void)out_size; (void)ws_size;

    float* outLogits = (float*)d_out;                       // [B,T,V]
    float* outAlpha  = outLogits + (size_t)B * T * V;       // [B,T,P]

    char* ws = (char*)d_ws;
    size_t off = 0;
    auto alloc = [&](size_t nfloats) {
        float* p = (float*)(ws + off);
        off += ((nfloats * sizeof(float) + 255) / 256) * 256;
        return p;
    };
    float* att1 = alloc((size_t)B * P * A_DIM);   // 25.7 MB
    float* Wcat = alloc((size_t)G4H * KCAT);      // 25.2 MB
    float* bcat = alloc(G4H);
    float* emb  = alloc((size_t)B * T * E);
    float* hbuf = alloc((size_t)B * H);           // hbuf,cbuf contiguous
    float* cbuf = alloc((size_t)B * H);
    float* att2 = alloc((size_t)B * A_DIM);
    float* ebuf = alloc((size_t)B * P);
    float* xbuf = alloc((size_t)B * KCAT);
    float* gbuf = alloc((size_t)B * G4H);

    auto gemm = [&](const float* Amat, const float* Wm, const float* bias, float* C,
                    int M, int N, int K, int ldc) {
        int nTiles = N / 16, nGroups = (nTiles + 3) / 4, mTiles = M / 16;
        dim3 grid((nGroups + 7) / 8, mTiles);
        gemm_wmma<<<grid, 256, 0, stream>>>(Amat, Wm, bias, C, M, N, K, ldc);
    };

    // ---- one-time (per call) prep
    zero_kernel<<<(2 * B * H + 255) / 256, 256, 0, stream>>>(hbuf, 2 * B * H); // h and c
    emb_gather<<<(B * T * E + 255) / 256, 256, 0, stream>>>(embW, caps, emb);
    build_wcat<<<(G4H * KCAT + 255) / 256, 256, 0, stream>>>(Wih, Whh, bih, bhh, Wcat, bcat);
    // att1 = feats @ We^T + be : [12544, 512], K=2048
    gemm(feats, We, be, att1, B * P, A_DIM, ENC, A_DIM);

    // ---- recurrent steps
    for (int t = 0; t < T; ++t) {
        // att2 = h @ Wd^T + bd : [64, 512]
        gemm(hbuf, Wd, bd, att2, B, A_DIM, H, A_DIM);
        attn_score<<<(B * P + 7) / 8, 256, 0, stream>>>(att1, att2, wf, bf, ebuf);
        softmax_ctx<<<B, 256, 0, stream>>>(ebuf, feats, xbuf, outAlpha, t);
        xfill<<<(B * (E + H) + 255) / 256, 256, 0, stream>>>(emb, hbuf, xbuf, t);
        // gates = [ctx|emb|h] @ Wcat^T + (b_ih+b_hh) : [64, 2048], K=3072
        gemm(xbuf, Wcat, bcat, gbuf, B, G4H, KCAT, G4H);
        lstm_cell<<<(B * H + 255) / 256, 256, 0, stream>>>(gbuf, hbuf, cbuf);
        // logits_t = h @ fc_W^T + fc_b, written straight into d_out at [:, t, :]
        gemm(hbuf, fcW, fcb, outLogits + (size_t)t * V, B, V, H, T * V);
    }
}